// TrajectoryTransformerVQVAE_59691455480154
// MI455X (gfx1250) — compile-verified
//
#include <hip/hip_runtime.h>
#include <hip/hip_bf16.h>
#include <math.h>

typedef __attribute__((ext_vector_type(2))) float v2f;
typedef __attribute__((ext_vector_type(8))) float v8f;
typedef __attribute__((ext_vector_type(4))) unsigned int u32x4;
typedef __attribute__((ext_vector_type(4))) int i32x4;
typedef __attribute__((ext_vector_type(8))) int i32x8;

// Tensor Data Mover availability (arity differs between toolchains).
#if defined(__has_builtin)
#if __has_builtin(__builtin_amdgcn_tensor_load_to_lds) && __has_builtin(__builtin_amdgcn_s_wait_tensorcnt)
#define TDM_OK 1
#endif
#endif
#ifndef TDM_OK
#define TDM_OK 0
#endif

// ---------------------------------------------------------------------------
// Embed: emb[b,p,:] = concat(states,actions) @ We^T + be  (p<60), eos (p==60),
// plus positional encoding. Also zeroes the qloss accumulator.
// ---------------------------------------------------------------------------
__global__ void embed_kernel(const float* __restrict__ states,
                             const float* __restrict__ actions,
                             const float* __restrict__ eos,
                             const float* __restrict__ pos,
                             const float* __restrict__ We,
                             const float* __restrict__ be,
                             float* __restrict__ emb,
                             float* __restrict__ qloss) {
  int idx = blockIdx.x * blockDim.x + threadIdx.x;
  if (idx == 0) *qloss = 0.0f;
  if (idx >= 128 * 61 * 8) return;
  int o = idx & 7;
  int p = (idx >> 3) % 61;
  int b = idx / (61 * 8);
  float v;
  if (p < 60) {
    v = be[o];
    const float* s = states + (b * 60 + p) * 17;
    const float* a = actions + (b * 60 + p) * 6;
    #pragma unroll
    for (int i = 0; i < 17; ++i) v += s[i] * We[o * 23 + i];
    #pragma unroll
    for (int i = 0; i < 6; ++i) v += a[i] * We[o * 23 + 17 + i];
  } else {
    v = eos[o];
  }
  v += pos[p * 8 + o];
  emb[idx] = v;
}

// ---------------------------------------------------------------------------
// Fused MHA + residual + LayerNorm.  D=8, H=4, dh=2, T<=61.
// mask modes: 0=none, 1=causal(k>q), 2=key-limit(k>tlim), 3=causal|key-limit
// out = LN(qin + MHA(qin, kvin))
// ---------------------------------------------------------------------------
__global__ __launch_bounds__(256) void attn_ln_kernel(
    const float* __restrict__ qin, const float* __restrict__ kvin,
    int Tq, int Tk, int mode, int tlim,
    const float* __restrict__ Wqkv, const float* __restrict__ bqkv,
    const float* __restrict__ Wo, const float* __restrict__ bo,
    const float* __restrict__ lng, const float* __restrict__ lnb,
    float* __restrict__ out) {
  __shared__ float sQ[64 * 8];
  __shared__ float sK[64 * 8];
  __shared__ float sV[64 * 8];
  __shared__ float sO[64 * 8];
  const int b = blockIdx.x;
  const float* qb = qin + b * Tq * 8;
  const float* kb = kvin + b * Tk * 8;
  float* ob = out + b * Tq * 8;

  for (int idx = threadIdx.x; idx < Tq * 8; idx += 256) {
    int t = idx >> 3, o = idx & 7;
    float s = bqkv[o];
    #pragma unroll
    for (int i = 0; i < 8; ++i) s += qb[t * 8 + i] * Wqkv[o * 8 + i];
    sQ[idx] = s;
  }
  for (int idx = threadIdx.x; idx < Tk * 8; idx += 256) {
    int t = idx >> 3, o = idx & 7;
    float sk = bqkv[8 + o], sv = bqkv[16 + o];
    #pragma unroll
    for (int i = 0; i < 8; ++i) {
      float x = kb[t * 8 + i];
      sk += x * Wqkv[(8 + o) * 8 + i];
      sv += x * Wqkv[(16 + o) * 8 + i];
    }
    sK[idx] = sk;
    sV[idx] = sv;
  }
  __syncthreads();
  for (int idx = threadIdx.x; idx < Tq * 4; idx += 256) {
    int q = idx >> 2, h = idx & 3;
    float q0 = sQ[q * 8 + h * 2], q1 = sQ[q * 8 + h * 2 + 1];
    float m = -3.0e38f, l = 0.0f, o0 = 0.0f, o1 = 0.0f;
    for (int k = 0; k < Tk; ++k) {
      float s = (q0 * sK[k * 8 + h * 2] + q1 * sK[k * 8 + h * 2 + 1]) * 0.70710678118f;
      bool msk = (mode == 1) ? (k > q)
               : (mode == 2) ? (k > tlim)
               : (mode == 3) ? ((k > q) || (k > tlim)) : false;
      if (msk) s = -1.0e9f;
      float nm = fmaxf(m, s);
      float sc = __expf(m - nm);
      float p = __expf(s - nm);
      l = l * sc + p;
      o0 = o0 * sc + p * sV[k * 8 + h * 2];
      o1 = o1 * sc + p * sV[k * 8 + h * 2 + 1];
      m = nm;
    }
    float inv = 1.0f / l;
    sO[q * 8 + h * 2] = o0 * inv;
    sO[q * 8 + h * 2 + 1] = o1 * inv;
  }
  __syncthreads();
  for (int t = threadIdx.x; t < Tq; t += 256) {
    float y[8];
    float mean = 0.0f;
    #pragma unroll
    for (int o = 0; o < 8; ++o) {
      float s = bo[o];
      #pragma unroll
      for (int i = 0; i < 8; ++i) s += sO[t * 8 + i] * Wo[o * 8 + i];
      s += qb[t * 8 + o];
      y[o] = s;
      mean += s;
    }
    mean *= 0.125f;
    float var = 0.0f;
    #pragma unroll
    for (int o = 0; o < 8; ++o) { float d = y[o] - mean; var += d * d; }
    var *= 0.125f;
    float rs = rsqrtf(var + 1e-5f);
    #pragma unroll
    for (int o = 0; o < 8; ++o) ob[t * 8 + o] = (y[o] - mean) * rs * lng[o] + lnb[o];
  }
}

#if TDM_OK
// 1-D bulk copy global -> LDS through the Tensor Data Mover.
// D# group0: count=1 | lds_addr | global_addr | type=2("image").
// D# group1: data_size=8B, tensor_dim0 = tile_dim0 = n (8-byte elems),
//            tensor_dim1 = tile_dim1 = 1, strides = n.  Groups 2/3 unused.
__device__ __forceinline__ void tdm_load_lds_1d(const void* gsrc, unsigned lds_off,
                                                unsigned nbytes) {
  unsigned long long ga = (unsigned long long)(uintptr_t)gsrc;
  unsigned n = nbytes >> 3;  // 8-byte elements
  u32x4 g0;
  g0[0] = 1u;                                            // count=1 (valid)
  g0[1] = lds_off;                                       // lds_addr
  g0[2] = (unsigned)(ga & 0xFFFFFFFFull);                // global_addr[31:0]
  g0[3] = (unsigned)((ga >> 32) & 0x1FFFFFFull) | (2u << 30);  // addr[56:32]|type
  i32x8 g1;
  g1[0] = (int)(3u << 16);                    // workgroup_mask=0, data_size=8B
  g1[1] = (int)((n & 0xFFFFu) << 16);         // tensor_dim0[15:0]
  g1[2] = (int)(((n >> 16) & 0xFFFFu) | (1u << 16));  // dim0[31:16], dim1=1
  g1[3] = (int)((n & 0xFFFFu) << 16);         // tile_dim0 = n
  g1[4] = 1;                                  // tile_dim1 = 1, tile_dim2 = 0
  g1[5] = (int)n;                             // tensor_dim0_stride[31:0]
  g1[6] = (int)((n & 0xFFFFu) << 16);         // stride0 hi=0, stride1[15:0]
  g1[7] = (int)(n >> 16);                     // stride1[47:16]
  i32x4 z4 = {0, 0, 0, 0};
#if __clang_major__ >= 23
  i32x8 z8 = {0, 0, 0, 0, 0, 0, 0, 0};
  __builtin_amdgcn_tensor_load_to_lds(g0, g1, z4, z4, z8, 0);
#else
  __builtin_amdgcn_tensor_load_to_lds(g0, g1, z4, z4, 0);
#endif
}
#endif

// ---------------------------------------------------------------------------
// Fused FF block + residual + LayerNorm via fp32 WMMA (V_WMMA_F32_16X16X4_F32).
//   h = relu(x @ W1^T + b1) : 128 N-tiles of 16, K=8 as 2 chained K=4 WMMAs
//   y = h @ W2^T + b2       : K=2048, corner-turn D->A layout through LDS
//   out = LN(x + y)
// 256 threads = 8 waves: wave = (tile 0..3) x (khalf 0..1); each (tile,khalf)
// accumulates 64 of the 128 hidden tiles; partials combined through LDS.
// W1/W2 (128 KB) staged once per block into dynamic LDS by the TDM.
// ---------------------------------------------------------------------------
__global__ __launch_bounds__(256) void ff_ln_kernel(
    const float* __restrict__ xin, int Ttot,
    const float* __restrict__ W1, const float* __restrict__ b1,
    const float* __restrict__ W2, const float* __restrict__ b2,
    const float* __restrict__ lng, const float* __restrict__ lnb,
    float* __restrict__ out) {
  extern __shared__ float dynsmem[];
  float* sCT = dynsmem;  // 8 waves * 256 floats: corner-turn + partial tiles
  const int b = blockIdx.x;
  const int lane = threadIdx.x & 31;
  const int wave = threadIdx.x >> 5;
  const int tile = wave & 3;
  const int khalf = wave >> 2;
  const float* xb = xin + b * Ttot * 8;
  float* ob = out + b * Ttot * 8;
  float* sc = sCT + wave * 256;

  const int lrow = lane & 15;
  const bool hi = lane >= 16;
  const int tok = tile * 16 + lrow;
  const bool tvalid = tok < Ttot;

#if TDM_OK
  float* sW1 = dynsmem + 8 * 256;
  float* sW2 = sW1 + 2048 * 8;
  if (wave == 0) {
    tdm_load_lds_1d(W1, (unsigned)(uintptr_t)(void*)sW1, 2048 * 8 * 4);
    tdm_load_lds_1d(W2, (unsigned)(uintptr_t)(void*)sW2, 8 * 2048 * 4);
    __builtin_amdgcn_s_wait_tensorcnt((short)0);
  }
  __syncthreads();
  const float* w1s = sW1;
  const float* w2s = sW2;
#else
  const float* w1s = W1;
  const float* w2s = W2;
#endif

  // A1 operands: 16x4 fp32 A-layout (VGPR0: K0|K2, VGPR1: K1|K3 across halves)
  v2f a1lo, a1hi;
  {
    int k0 = hi ? 2 : 0;
    a1lo.x = tvalid ? xb[tok * 8 + k0] : 0.0f;
    a1lo.y = tvalid ? xb[tok * 8 + k0 + 1] : 0.0f;
    a1hi.x = tvalid ? xb[tok * 8 + k0 + 4] : 0.0f;
    a1hi.y = tvalid ? xb[tok * 8 + k0 + 5] : 0.0f;
  }

  v8f acc;
  #pragma unroll
  for (int j = 0; j < 8; ++j) acc[j] = 0.0f;

  const int kb = hi ? 1 : 0;
  const int nt0 = khalf * 64;
  for (int i = 0; i < 64; ++i) {
    const int nt = nt0 + i;
    __builtin_prefetch(&b1[(nt + 1) * 16], 0, 3);
    const int n = nt * 16 + lrow;
    // B1 chunk of W1^T: B[4x16], rows K striped across lane halves
    v2f b1lo, b1hi;
    b1lo.x = w1s[n * 8 + kb];
    b1lo.y = w1s[n * 8 + kb + 2];
    b1hi.x = w1s[n * 8 + 4 + kb];
    b1hi.y = w1s[n * 8 + 4 + kb + 2];
    // C preloaded with bias b1 (each lane's column is n = lrow)
    float bias1 = b1[n];
    v8f c;
    #pragma unroll
    for (int j = 0; j < 8; ++j) c[j] = bias1;
    c = __builtin_amdgcn_wmma_f32_16x16x4_f32(false, a1lo, false, b1lo, (short)0, c, false, false);
    c = __builtin_amdgcn_wmma_f32_16x16x4_f32(false, a1hi, false, b1hi, (short)0, c, false, false);
    #pragma unroll
    for (int j = 0; j < 8; ++j) c[j] = fmaxf(c[j], 0.0f);

    // corner turn: D-layout -> row-major [token][hid] in per-wave LDS tile
    #pragma unroll
    for (int r = 0; r < 8; ++r) sc[(hi ? r + 8 : r) * 16 + lrow] = c[r];
    __asm__ volatile("s_wait_dscnt 0" ::: "memory");

    // FF2: K = 16 local hidden dims -> 4 chained K=4 WMMAs
    #pragma unroll
    for (int kk = 0; kk < 4; ++kk) {
      v2f a2;
      int kbase = kk * 4 + (hi ? 2 : 0);
      a2.x = sc[lrow * 16 + kbase];
      a2.y = sc[lrow * 16 + kbase + 1];
      v2f b2v;
      int kg = nt * 16 + kk * 4 + kb;
      b2v.x = (lrow < 8) ? w2s[lrow * 2048 + kg] : 0.0f;
      b2v.y = (lrow < 8) ? w2s[lrow * 2048 + kg + 2] : 0.0f;
      acc = __builtin_amdgcn_wmma_f32_16x16x4_f32(false, a2, false, b2v, (short)0, acc, false, false);
    }
    __asm__ volatile("s_wait_dscnt 0" ::: "memory");  // WAR on sc before next nt
  }

  // publish partial accumulator tile, combine khalf pairs, then LN epilogue
  #pragma unroll
  for (int r = 0; r < 8; ++r) sc[(hi ? r + 8 : r) * 16 + lrow] = acc[r];
  __syncthreads();
  if (khalf == 0 && lane < 16 && tok < Ttot) {
    float y[8];
    float mean = 0.0f;
    #pragma unroll
    for (int o = 0; o < 8; ++o) {
      float v = sCT[wave * 256 + lane * 16 + o] + sCT[(wave + 4) * 256 + lane * 16 + o] +
                b2[o] + xb[tok * 8 + o];
      y[o] = v;
      mean += v;
    }
    mean *= 0.125f;
    float var = 0.0f;
    #pragma unroll
    for (int o = 0; o < 8; ++o) { float d = y[o] - mean; var += d * d; }
    var *= 0.125f;
    float rs = rsqrtf(var + 1e-5f);
    #pragma unroll
    for (int o = 0; o < 8; ++o) ob[tok * 8 + o] = (y[o] - mean) * rs * lng[o] + lnb[o];
  }
}

// ---------------------------------------------------------------------------
// Plain LayerNorm over last dim (8), one thread per token.
// ---------------------------------------------------------------------------
__global__ void ln_kernel(const float* __restrict__ x,
                          const float* __restrict__ g,
                          const float* __restrict__ bb,
                          float* __restrict__ out, int ntok) {
  int t = blockIdx.x * blockDim.x + threadIdx.x;
  if (t >= ntok) return;
  float y[8];
  float mean = 0.0f;
  #pragma unroll
  for (int o = 0; o < 8; ++o) { y[o] = x[t * 8 + o]; mean += y[o]; }
  mean *= 0.125f;
  float var = 0.0f;
  #pragma unroll
  for (int o = 0; o < 8; ++o) { float d = y[o] - mean; var += d * d; }
  var *= 0.125f;
  float rs = rsqrtf(var + 1e-5f);
  #pragma unroll
  for (int o = 0; o < 8; ++o) out[t * 8 + o] = (y[o] - mean) * rs * g[o] + bb[o];
}

// ---------------------------------------------------------------------------
// VQ: eos = encoded[:, 60, :]; nearest of 15 codes; qloss += (1+beta)*d/(B*D);
// init decode buffer buf[:,0,:]=eos, rest zeros.
// ---------------------------------------------------------------------------
__global__ __launch_bounds__(64) void vq_kernel(const float* __restrict__ encoded,
                                                const float* __restrict__ codebook,
                                                float* __restrict__ buf,
                                                float* __restrict__ qloss) {
  __shared__ float eosv[8];
  int b = blockIdx.x;
  if (threadIdx.x < 8) eosv[threadIdx.x] = encoded[(b * 61 + 60) * 8 + threadIdx.x];
  __syncthreads();
  if (threadIdx.x == 0) {
    float best = 3.0e38f;
    for (int c = 0; c < 15; ++c) {
      float d = 0.0f;
      #pragma unroll
      for (int i = 0; i < 8; ++i) {
        float df = codebook[c * 8 + i] - eosv[i];
        d += df * df;
      }
      best = fminf(best, d);
    }
    atomicAdd(qloss, best * (1.0f + 0.01f) / (128.0f * 8.0f));
  }
  for (int i = threadIdx.x; i < 60 * 8; i += 64)
    buf[b * 60 * 8 + i] = (i < 8) ? eosv[i] : 0.0f;
}

// ---------------------------------------------------------------------------
// Head: next_state = out[:,t,:] @ Wfc^T + bfc -> predicted[:,t,:];
// buf[:, min(t+1,59), :] = next_state @ Wse^T + bse
// ---------------------------------------------------------------------------
__global__ __launch_bounds__(32) void head_kernel(const float* __restrict__ dec_out, int t,
                                                  const float* __restrict__ Wfc,
                                                  const float* __restrict__ bfc,
                                                  const float* __restrict__ Wse,
                                                  const float* __restrict__ bse,
                                                  float* __restrict__ pred,
                                                  float* __restrict__ buf) {
  __shared__ float ns[17];
  int b = blockIdx.x;
  const float* h = dec_out + (b * 60 + t) * 8;
  if (threadIdx.x < 17) {
    float s = bfc[threadIdx.x];
    #pragma unroll
    for (int i = 0; i < 8; ++i) s += h[i] * Wfc[threadIdx.x * 8 + i];
    ns[threadIdx.x] = s;
    pred[(b * 60 + t) * 17 + threadIdx.x] = s;
  }
  __syncthreads();
  int tn = (t + 1 < 60) ? (t + 1) : 59;
  if (threadIdx.x < 8) {
    float s = bse[threadIdx.x];
    #pragma unroll
    for (int i = 0; i < 17; ++i) s += ns[i] * Wse[threadIdx.x * 17 + i];
    buf[(b * 60 + tn) * 8 + threadIdx.x] = s;
  }
}

// ---------------------------------------------------------------------------
// Host orchestration.  JAX pytree leaf order: top-level in insertion order;
// nested dicts sorted by key.  Transformer blob (n_dec dec, n_enc enc layers):
//   dec layer i  @ base+i*18 : W1,W2,b1,b2, caWo,caWqkv,cabo,cabqkv,
//                              ln1g,ln1b,ln2g,ln2b,ln3g,ln3b, saWo,saWqkv,sabo,sabqkv
//   dec_norm @ base+n_dec*18; enc layer i @ base+n_dec*18+2+i*12 :
//     W1,W2, aWo,aWqkv,abo,abqkv, b1,b2, ln1g,ln1b,ln2g,ln2b
//   enc_norm @ base+n_dec*18+2+n_enc*12
// ---------------------------------------------------------------------------
extern "C" void kernel_launch(void* const* d_in, const int* in_sizes, int n_in,
                              void* d_out, int out_size, void* d_ws, size_t ws_size,
                              hipStream_t stream) {
  (void)in_sizes; (void)n_in; (void)out_size; (void)ws_size;
  auto F = [&](int i) { return (const float*)d_in[i]; };

  float* ws = (float*)d_ws;
  float* e0 = ws;              // [128,61,8] embedding (preserved)
  float* ea = e0 + 62464;      // [128,61,8]
  float* eb = ea + 62464;      // [128,61,8]
  float* em = eb + 62464;      // [128,61,8] encoder "mem"
  float* da = em + 62464;      // [128,60,8]
  float* db = da + 61440;      // [128,60,8]
  float* dm = db + 61440;      // [128,60,8] decoder-phase "mem"
  float* buf = dm + 61440;     // [128,60,8] autoregressive state
  float* pred = (float*)d_out;           // [128,60,17]
  float* qloss = pred + 128 * 60 * 17;   // scalar

  const int EB = 6;    // enc_tf blob (n_dec=6, n_enc=2): leaves 6..141
  const int DB = 144;  // dec_tf blob (n_dec=2, n_enc=6): leaves 144..255
  const int WSE = 142, BSE = 143, WFC = 256, BFC = 257, CB = 258;
  const int EB_DECNORM = EB + 6 * 18;           // 114
  const int EB_ENC = EB + 6 * 18 + 2;           // 116
  const int EB_ENCNORM = EB_ENC + 2 * 12;       // 140
  const int DB_DECNORM = DB + 2 * 18;           // 180
  const int DB_ENC = DB + 2 * 18 + 2;           // 182
  const int DB_ENCNORM = DB_ENC + 6 * 12;       // 254

#if TDM_OK
  const size_t ffShmem = (8 * 256 + 2 * 2048 * 8) * sizeof(float);  // 136 KB
#else
  const size_t ffShmem = (8 * 256) * sizeof(float);                 // 8 KB
#endif

  auto attn = [&](const float* qi, const float* kvi, int Tq, int Tk, int mode, int tlim,
                  int p /* Wo,Wqkv,bo,bqkv */, int lg, int lb, float* o) {
    attn_ln_kernel<<<128, 256, 0, stream>>>(qi, kvi, Tq, Tk, mode, tlim,
                                            F(p + 1), F(p + 3), F(p + 0), F(p + 2),
                                            F(lg), F(lb), o);
  };
  auto ff = [&](const float* xi, int Ttot, int w1, int bb1, int w2, int bb2,
                int lg, int lb, float* o) {
    ff_ln_kernel<<<128, 256, ffShmem, stream>>>(xi, Ttot, F(w1), F(bb1), F(w2), F(bb2),
                                                F(lg), F(lb), o);
  };
  auto ln = [&](const float* xi, int Ttot, int gi, int bi, float* o) {
    int n = 128 * Ttot;
    ln_kernel<<<(n + 255) / 256, 256, 0, stream>>>(xi, F(gi), F(bi), o, n);
  };

  // ---- embed -------------------------------------------------------------
  embed_kernel<<<(128 * 61 * 8 + 255) / 256, 256, 0, stream>>>(
      F(0), F(1), F(2), F(3), F(4), F(5), e0, qloss);

  // ---- enc_tf: 2 enc layers (causal), mem = LN, 6 dec layers (no mask) ---
  {
    int L = EB_ENC + 0 * 12;
    attn(e0, e0, 61, 61, 1, 0, L + 2, L + 8, L + 9, ea);
    ff(ea, 61, L + 0, L + 6, L + 1, L + 7, L + 10, L + 11, eb);
    L = EB_ENC + 1 * 12;
    attn(eb, eb, 61, 61, 1, 0, L + 2, L + 8, L + 9, ea);
    ff(ea, 61, L + 0, L + 6, L + 1, L + 7, L + 10, L + 11, eb);
    ln(eb, 61, EB_ENCNORM, EB_ENCNORM + 1, em);
    const float* cur = e0;
    for (int i = 0; i < 6; ++i) {
      int D = EB + i * 18;
      attn(cur, cur, 61, 61, 0, 0, D + 14, D + 8, D + 9, ea);   // self-attn
      attn(ea, em, 61, 61, 0, 0, D + 4, D + 10, D + 11, ea);    // cross-attn
      ff(ea, 61, D + 0, D + 2, D + 1, D + 3, D + 12, D + 13, ea);
      cur = ea;
    }
    ln(ea, 61, EB_DECNORM, EB_DECNORM + 1, ea);  // encoded
  }

  // ---- VQ + buf init -----------------------------------------------------
  vq_kernel<<<128, 64, 0, stream>>>(ea, F(CB), buf, qloss);

  // ---- 60 sequential decode steps (dec_tf: 6 enc layers + 2 dec layers) --
  for (int t = 0; t < 60; ++t) {
    const float* cur = buf;
    for (int i = 0; i < 6; ++i) {
      int L = DB_ENC + i * 12;
      attn(cur, cur, 60, 60, 2, t, L + 2, L + 8, L + 9, da);  // key mask pos>t
      ff(da, 60, L + 0, L + 6, L + 1, L + 7, L + 10, L + 11, da);
      cur = da;
    }
    ln(da, 60, DB_ENCNORM, DB_ENCNORM + 1, dm);
    const float* y = buf;
    for (int i = 0; i < 2; ++i) {
      int D = DB + i * 18;
      attn(y, y, 60, 60, 3, t, D + 14, D + 8, D + 9, db);   // causal | pos>t
      attn(db, dm, 60, 60, 2, t, D + 4, D + 10, D + 11, db);
      ff(db, 60, D + 0, D + 2, D + 1, D + 3, D + 12, D + 13, db);
      y = db;
    }
    ln(db, 60, DB_DECNORM, DB_DECNORM + 1, db);
    head_kernel<<<128, 32, 0, stream>>>(db, t, F(WFC), F(BFC), F(WSE), F(BSE),
                                        pred, buf);
  }
}